// SpatialAttentionWithMemory_54589034332874
// MI455X (gfx1250) — compile-verified
//
#include <hip/hip_runtime.h>
#include <hip/hip_bf16.h>

// Problem constants (from reference setup_inputs)
#define B_  2
#define C_  64
#define H_  24
#define N_  13824          // 24*24*24
#define M_  5
#define NT_ 864            // N_/16 n-tiles
#define BCN ((size_t)B_ * C_ * N_)

typedef __attribute__((ext_vector_type(2))) float v2f;
typedef __attribute__((ext_vector_type(8))) float v8f;

__device__ __forceinline__ v8f wmma_f32(v2f a, v2f b, v8f c) {
    // V_WMMA_F32_16X16X4_F32 : D(16x16 f32) = A(16x4) * B(4x16) + C
    return __builtin_amdgcn_wmma_f32_16x16x4_f32(
        false, a, false, b, (short)0, c, false, false);
}

__device__ __forceinline__ float sigmoidf_(float x) {
    return 1.0f / (1.0f + __expf(-x));
}

// ---------------------------------------------------------------------------
// Kernel 1: q = Wq x + bq ; kx = Wk x ; vx = Wv x   (K=64 GEMM via WMMA f32)
// One wave -> one 16(o) x 16(n) tile, all three weight matrices fused.
// ---------------------------------------------------------------------------
__global__ void __launch_bounds__(256)
qkv_gemm_kernel(const float* __restrict__ x,
                const float* __restrict__ Wq, const float* __restrict__ bq,
                const float* __restrict__ Wk, const float* __restrict__ Wv,
                float* __restrict__ q, float* __restrict__ kx,
                float* __restrict__ vx) {
    const int wid  = threadIdx.x >> 5;
    const int lane = threadIdx.x & 31;
    const int half = lane >> 4;
    const int l16  = lane & 15;

    const int tile = blockIdx.x * 8 + wid;        // 0 .. B*NT*4-1
    const int b    = tile / (NT_ * 4);
    const int rem  = tile % (NT_ * 4);
    const int nt   = rem >> 2;
    const int ot   = rem & 3;

    const int o0 = ot * 16;
    const int n0 = nt * 16;
    const int n  = n0 + l16;
    const float* xb = x + (size_t)b * C_ * N_;
    const int arow  = o0 + l16;

    v8f accq = {}; v8f acck = {}; v8f accv = {};
#pragma unroll
    for (int ks = 0; ks < 16; ++ks) {
        const int ca = ks * 4 + 2 * half;         // K pair handled by this lane
        v2f aq = *(const v2f*)(Wq + arow * 64 + ca);
        v2f ak = *(const v2f*)(Wk + arow * 64 + ca);
        v2f av = *(const v2f*)(Wv + arow * 64 + ca);
        v2f bx;
        bx.x = xb[(size_t)ca * N_ + n];
        bx.y = xb[(size_t)(ca + 1) * N_ + n];
        accq = wmma_f32(aq, bx, accq);
        acck = wmma_f32(ak, bx, acck);
        accv = wmma_f32(av, bx, accv);
    }
#pragma unroll
    for (int r = 0; r < 8; ++r) {
        const int o = o0 + r + half * 8;
        const size_t off = ((size_t)b * C_ + o) * N_ + n;
        q[off]  = accq[r] + bq[o];
        kx[off] = acck[r];        // bias folded in during attention (padding!)
        vx[off] = accv[r];
    }
}

// ---------------------------------------------------------------------------
// Kernel 2: per-channel scalar attention over 5 memory tokens + 27 neighbors.
// Online softmax; neighbor k/v gathered from kx/vx (shift-commute with the
// 1x1x1 conv); zero-padded neighbors contribute exactly bk/bv.
// ---------------------------------------------------------------------------
__global__ void __launch_bounds__(256)
attn_kernel(const float* __restrict__ q,  const float* __restrict__ kx,
            const float* __restrict__ vx, const float* __restrict__ mem_key,
            const float* __restrict__ mem_value,
            const float* __restrict__ bk, const float* __restrict__ bv,
            float* __restrict__ attn_out) {
    const size_t idx = (size_t)blockIdx.x * blockDim.x + threadIdx.x;
    if (idx >= BCN) return;
    const int n  = (int)(idx % N_);
    const int bc = (int)(idx / N_);
    const int c  = bc % C_;

    const float qv = q[idx] * 0.125f;             // 1/sqrt(64)
    float m = -3.402823466e+38f, den = 0.0f, acc = 0.0f;

#pragma unroll
    for (int mi = 0; mi < M_; ++mi) {
        const float kkv = mem_key[c * M_ + mi];
        const float vvv = mem_value[c * M_ + mi];
        const float l   = qv * kkv;
        const float mn  = fmaxf(m, l);
        const float cr  = __expf(m - mn);
        const float e   = __expf(l - mn);
        den = den * cr + e;
        acc = acc * cr + e * vvv;
        m = mn;
    }

    const int h = n / 576, r2 = n % 576, d = r2 / 24, w = r2 % 24;
    const float* kb = kx + (size_t)bc * N_;
    const float* vb = vx + (size_t)bc * N_;
    const float bkc = bk[c], bvc = bv[c];

#pragma unroll
    for (int dh = -1; dh <= 1; ++dh)
#pragma unroll
        for (int dd = -1; dd <= 1; ++dd)
#pragma unroll
            for (int dw = -1; dw <= 1; ++dw) {
                const bool in = (unsigned)(h + dh) < 24u &&
                                (unsigned)(d + dd) < 24u &&
                                (unsigned)(w + dw) < 24u;
                float kkv = bkc, vvv = bvc;
                if (in) {
                    const int nn = n + dh * 576 + dd * 24 + dw;
                    kkv += kb[nn];
                    vvv += vb[nn];
                }
                const float l  = qv * kkv;
                const float mn = fmaxf(m, l);
                const float cr = __expf(m - mn);
                const float e  = __expf(l - mn);
                den = den * cr + e;
                acc = acc * cr + e * vvv;
                m = mn;
            }
    attn_out[idx] = acc / den;
}

// ---------------------------------------------------------------------------
// Kernel 3: gate = sigmoid(Wg [x; attn] + bg) via K=128 WMMA GEMM, fused with
// the gated blend: out = g*attn + (1-g)*x.
// ---------------------------------------------------------------------------
__global__ void __launch_bounds__(256)
gate_gemm_kernel(const float* __restrict__ x,  const float* __restrict__ at,
                 const float* __restrict__ Wg, const float* __restrict__ bg,
                 float* __restrict__ out) {
    const int wid  = threadIdx.x >> 5;
    const int lane = threadIdx.x & 31;
    const int half = lane >> 4;
    const int l16  = lane & 15;

    const int tile = blockIdx.x * 8 + wid;
    const int b    = tile / (NT_ * 4);
    const int rem  = tile % (NT_ * 4);
    const int nt   = rem >> 2;
    const int ot   = rem & 3;

    const int o0 = ot * 16;
    const int n0 = nt * 16;
    const int n  = n0 + l16;
    const int arow = o0 + l16;

    v8f acc = {};
#pragma unroll
    for (int ks = 0; ks < 32; ++ks) {
        const int ca = ks * 4 + 2 * half;         // 0..127, even -> never
        v2f a = *(const v2f*)(Wg + arow * 128 + ca);  //   straddles x/attn
        const float* cb0 = (ca < 64)
            ? (x  + ((size_t)b * C_ + ca)        * N_)
            : (at + ((size_t)b * C_ + (ca - 64)) * N_);
        v2f bx;
        bx.x = cb0[n];
        bx.y = cb0[N_ + n];
        acc = wmma_f32(a, bx, acc);
    }
#pragma unroll
    for (int r = 0; r < 8; ++r) {
        const int o = o0 + r + half * 8;
        const size_t off = ((size_t)b * C_ + o) * N_ + n;
        const float g  = sigmoidf_(acc[r] + bg[o]);
        const float av = at[off];
        const float xv = x[off];
        out[off] = g * av + (1.0f - g) * xv;
    }
}

// ---------------------------------------------------------------------------
// Kernel 4: deterministic spatial mean per (b,c) (fixed-order tree reduction,
// no float atomics -> bit-identical across graph replays).
// ---------------------------------------------------------------------------
__global__ void __launch_bounds__(256)
mean_kernel(const float* __restrict__ out, float* __restrict__ mu) {
    const int bc = blockIdx.x;
    const float* p = out + (size_t)bc * N_;
    float s = 0.0f;
    for (int i = threadIdx.x; i < N_; i += 256) s += p[i];
    __shared__ float sm[256];
    sm[threadIdx.x] = s;
    __syncthreads();
    for (int st = 128; st > 0; st >>= 1) {
        if (threadIdx.x < st) sm[threadIdx.x] += sm[threadIdx.x + st];
        __syncthreads();
    }
    if (threadIdx.x == 0) mu[bc] = sm[0] * (1.0f / (float)N_);
}

// ---------------------------------------------------------------------------
// Kernel 5: GRU cell on pooled features (tiny: 2*64 states, K=64 dots).
// ---------------------------------------------------------------------------
__global__ void gru_kernel(const float* __restrict__ mu,
                           const float* __restrict__ prev,
                           const float* __restrict__ W_ih,
                           const float* __restrict__ W_hh,
                           const float* __restrict__ b_ih,
                           const float* __restrict__ b_hh,
                           float* __restrict__ newmem) {
    const int t = threadIdx.x;
    if (t >= B_ * C_) return;
    const int b = t >> 6, c = t & 63;
    const float* mb = mu   + b * C_;
    const float* pb = prev + b * C_;
    float ir = b_ih[c], iz = b_ih[64 + c], ii = b_ih[128 + c];
    float hr = b_hh[c], hz = b_hh[64 + c], hn = b_hh[128 + c];
    for (int j = 0; j < 64; ++j) {
        const float mv = mb[j], pv = pb[j];
        ir += W_ih[c * 64 + j] * mv;
        iz += W_ih[(64 + c) * 64 + j] * mv;
        ii += W_ih[(128 + c) * 64 + j] * mv;
        hr += W_hh[c * 64 + j] * pv;
        hz += W_hh[(64 + c) * 64 + j] * pv;
        hn += W_hh[(128 + c) * 64 + j] * pv;
    }
    const float r  = sigmoidf_(ir + hr);
    const float z  = sigmoidf_(iz + hz);
    const float ng = tanhf(ii + r * hn);
    newmem[t] = (1.0f - z) * ng + z * pb[c];
}

// ---------------------------------------------------------------------------
extern "C" void kernel_launch(void* const* d_in, const int* in_sizes, int n_in,
                              void* d_out, int out_size, void* d_ws,
                              size_t ws_size, hipStream_t stream) {
    (void)in_sizes; (void)n_in; (void)out_size; (void)ws_size;

    const float* x         = (const float*)d_in[0];
    const float* prev_mem  = (const float*)d_in[1];
    const float* Wq        = (const float*)d_in[2];
    const float* bq        = (const float*)d_in[3];
    const float* Wk        = (const float*)d_in[4];
    const float* bk        = (const float*)d_in[5];
    const float* Wv        = (const float*)d_in[6];
    const float* bv        = (const float*)d_in[7];
    const float* mem_key   = (const float*)d_in[8];
    const float* mem_value = (const float*)d_in[9];
    const float* Wg        = (const float*)d_in[10];
    const float* bg        = (const float*)d_in[11];
    const float* W_ih      = (const float*)d_in[12];
    const float* W_hh      = (const float*)d_in[13];
    const float* b_ih      = (const float*)d_in[14];
    const float* b_hh      = (const float*)d_in[15];

    float* out    = (float*)d_out;          // (B,C,H,D,W) flat
    float* newmem = out + BCN;              // (B,C) appended

    float* ws = (float*)d_ws;
    float* q  = ws;                         // BCN
    float* kx = q  + BCN;                   // BCN
    float* vx = kx + BCN;                   // BCN
    float* at = vx + BCN;                   // BCN
    float* mu = at + BCN;                   // B*C

    // 1) fused q/k/v channel-mix GEMM (B*NT*4 tiles, 8 waves/block)
    qkv_gemm_kernel<<<(B_ * NT_ * 4) / 8, 256, 0, stream>>>(
        x, Wq, bq, Wk, Wv, q, kx, vx);

    // 2) per-channel online-softmax attention
    attn_kernel<<<(unsigned)(BCN / 256), 256, 0, stream>>>(
        q, kx, vx, mem_key, mem_value, bk, bv, at);

    // 3) gate GEMM (K=128) + gated blend -> out
    gate_gemm_kernel<<<(B_ * NT_ * 4) / 8, 256, 0, stream>>>(
        x, at, Wg, bg, out);

    // 4) deterministic spatial mean
    mean_kernel<<<B_ * C_, 256, 0, stream>>>(out, mu);

    // 5) GRU memory update
    gru_kernel<<<1, 128, 0, stream>>>(mu, prev_mem, W_ih, W_hh, b_ih, b_hh,
                                      newmem);
}